// ReservoirRNNCell_65541200937001
// MI455X (gfx1250) — compile-verified
//
#include <hip/hip_runtime.h>
#include <hip/hip_bf16.h>

// Problem constants (from reference)
#define BB   128
#define IN_  1024
#define HH   1024
// TAU == 1.0 -> both /TAU are identity
#define FW_DECAY 0.01f
#define FW_LR    0.01f

typedef __attribute__((ext_vector_type(2))) float v2f;
typedef __attribute__((ext_vector_type(8))) float v8f;

// ---------------------------------------------------------------------------
// K1: logits = W_hh_fixed + W_hh_virtual  (H*H, float4 vectorized)
// ---------------------------------------------------------------------------
__global__ __launch_bounds__(256) void k_add_logits(const float* __restrict__ a,
                                                    const float* __restrict__ b,
                                                    float* __restrict__ o) {
    int i = blockIdx.x * blockDim.x + threadIdx.x;        // i indexes float4
    float4 va = reinterpret_cast<const float4*>(a)[i];
    float4 vb = reinterpret_cast<const float4*>(b)[i];
    float4 vo;
    vo.x = va.x + vb.x; vo.y = va.y + vb.y;
    vo.z = va.z + vb.z; vo.w = va.w + vb.w;
    reinterpret_cast<float4*>(o)[i] = vo;
}

// ---------------------------------------------------------------------------
// K2: input_contrib[b,o] = sum_k x_t[b,k] * W_ih[o,k] + b_ih[o]
// One wave per 16x16 output tile, V_WMMA_F32_16X16X4_F32 over K=1024.
// A frag (16x4 f32): lane m=l%16; VGPR0/1 hold K = (l<16?0:2)+{0,1}
// C/D (16x16 f32): vgpr r -> M = r + (l<16?0:8), N = l%16
// ---------------------------------------------------------------------------
__global__ __launch_bounds__(32) void k_gemm_input(const float* __restrict__ x,
                                                   const float* __restrict__ Wih,
                                                   const float* __restrict__ bih,
                                                   float* __restrict__ out) {
    const int lane  = threadIdx.x;
    const int b0    = blockIdx.x * 16;
    const int o0    = blockIdx.y * 16;
    const int row   = lane & 15;
    const int khalf = (lane >> 4) << 1;                   // 0 or 2

    const float* __restrict__ arow = x   + (size_t)(b0 + row) * IN_;
    const float* __restrict__ brow = Wih + (size_t)(o0 + row) * IN_;

    v8f acc = {};
    for (int k0 = 0; k0 < IN_; k0 += 4) {
        float2 af = *reinterpret_cast<const float2*>(arow + k0 + khalf);
        float2 bf = *reinterpret_cast<const float2*>(brow + k0 + khalf);
        v2f av; av[0] = af.x; av[1] = af.y;
        v2f bv; bv[0] = bf.x; bv[1] = bf.y;
        acc = __builtin_amdgcn_wmma_f32_16x16x4_f32(
            /*neg_a=*/false, av, /*neg_b=*/false, bv,
            /*c_mod=*/(short)0, acc, /*reuse_a=*/false, /*reuse_b=*/false);
    }

    const int n     = o0 + row;
    const int mbase = b0 + ((lane >> 4) << 3);
    const float bias = bih[n];
#pragma unroll
    for (int r = 0; r < 8; ++r)
        out[(size_t)(mbase + r) * HH + n] = acc[r] + bias;
}

// ---------------------------------------------------------------------------
// K3 (hot): fused Gumbel-softmax exchange + tanh.
// One wave per (b,o) row of 1024. 8 waves/block share b -> h_prev in LDS.
// contrib = (sum_h e^{z-m} * h_prev[h]) / (sum_h e^{z-m}),
//   z = logits[o,h] - log(-log(clip(u)))
// ---------------------------------------------------------------------------
__device__ __forceinline__ float gumbel_z(float u, float lg) {
    float uc = fminf(fmaxf(u, 1e-9f), 1.0f - 1e-9f);
    return lg - __logf(-__logf(uc));
}

__global__ __launch_bounds__(256) void k_fused_exchange(
        const float* __restrict__ gumbel, const float* __restrict__ h_prev,
        const float* __restrict__ logits, const float* __restrict__ input_contrib,
        float* __restrict__ h_next) {
    __shared__ float sh_h[HH];                            // 4 KB: h_prev[b,:]

    const int tid  = threadIdx.x;
    const int lane = tid & 31;
    const int wave = tid >> 5;
    const int row0 = blockIdx.x * 8;                      // 8 rows per block
    const int b    = row0 / HH;                           // all 8 share b (8 | H)
    const int o    = (row0 + wave) % HH;

    // cooperative LDS stage of h_prev[b,:]
    reinterpret_cast<float4*>(sh_h)[tid] =
        reinterpret_cast<const float4*>(h_prev + (size_t)b * HH)[tid];
    __syncthreads();

    const float* __restrict__ grow = gumbel + ((size_t)b * HH + o) * HH;
    const float* __restrict__ lrow = logits + (size_t)o * HH;

    // pass 1: z into registers, running max
    float4 z[8];
    float m = -INFINITY;
#pragma unroll
    for (int c = 0; c < 8; ++c) {
        const int h = c * 128 + lane * 4;
        float4 u  = *reinterpret_cast<const float4*>(grow + h);
        float4 lg = *reinterpret_cast<const float4*>(lrow + h);
        float4 zz;
        zz.x = gumbel_z(u.x, lg.x);
        zz.y = gumbel_z(u.y, lg.y);
        zz.z = gumbel_z(u.z, lg.z);
        zz.w = gumbel_z(u.w, lg.w);
        m = fmaxf(m, fmaxf(fmaxf(zz.x, zz.y), fmaxf(zz.z, zz.w)));
        z[c] = zz;
    }
    // wave32 max-reduce
#pragma unroll
    for (int off = 16; off > 0; off >>= 1)
        m = fmaxf(m, __shfl_xor(m, off, 32));

    // pass 2: exp + weighted sums (h_prev from LDS)
    float s0 = 0.0f, s1 = 0.0f;
#pragma unroll
    for (int c = 0; c < 8; ++c) {
        const int h = c * 128 + lane * 4;
        float4 hv = *reinterpret_cast<const float4*>(sh_h + h);
        float e;
        e = __expf(z[c].x - m); s0 += e; s1 += e * hv.x;
        e = __expf(z[c].y - m); s0 += e; s1 += e * hv.y;
        e = __expf(z[c].z - m); s0 += e; s1 += e * hv.z;
        e = __expf(z[c].w - m); s0 += e; s1 += e * hv.w;
    }
#pragma unroll
    for (int off = 16; off > 0; off >>= 1) {
        s0 += __shfl_xor(s0, off, 32);
        s1 += __shfl_xor(s1, off, 32);
    }

    if (lane == 0) {
        const float contrib = s1 / s0;
        const size_t idx = (size_t)b * HH + o;
        h_next[idx] = tanhf(input_contrib[idx] + contrib);
    }
}

// ---------------------------------------------------------------------------
// K4: W_new[o,h] = 0.99*W_virtual[o,h] + (0.01/B) * sum_b h_next[b,o]*h_prev[b,h]
// One wave per 16x16 tile, WMMA f32 over K = B = 128.
// A[m,k] = h_next[k, o0+m] ; B[k,n] = h_prev[k, h0+n]  (column gathers)
// ---------------------------------------------------------------------------
__global__ __launch_bounds__(32) void k_gemm_hebbian(const float* __restrict__ h_next,
                                                     const float* __restrict__ h_prev,
                                                     const float* __restrict__ Wv,
                                                     float* __restrict__ outW) {
    const int lane  = threadIdx.x;
    const int o0    = blockIdx.x * 16;
    const int h0    = blockIdx.y * 16;
    const int col   = lane & 15;
    const int khalf = (lane >> 4) << 1;

    v8f acc = {};
    for (int k0 = 0; k0 < BB; k0 += 4) {
        const int k = k0 + khalf;
        v2f av, bv;
        av[0] = h_next[(size_t)(k    ) * HH + o0 + col];
        av[1] = h_next[(size_t)(k + 1) * HH + o0 + col];
        bv[0] = h_prev[(size_t)(k    ) * HH + h0 + col];
        bv[1] = h_prev[(size_t)(k + 1) * HH + h0 + col];
        acc = __builtin_amdgcn_wmma_f32_16x16x4_f32(
            false, av, false, bv, (short)0, acc, false, false);
    }

    const int n     = h0 + col;
    const int mbase = o0 + ((lane >> 4) << 3);
    const float scale = FW_LR / (float)BB;
#pragma unroll
    for (int r = 0; r < 8; ++r) {
        const size_t idx = (size_t)(mbase + r) * HH + n;
        outW[idx] = Wv[idx] * (1.0f - FW_DECAY) + acc[r] * scale;
    }
}

// ---------------------------------------------------------------------------
extern "C" void kernel_launch(void* const* d_in, const int* in_sizes, int n_in,
                              void* d_out, int out_size, void* d_ws, size_t ws_size,
                              hipStream_t stream) {
    (void)in_sizes; (void)n_in; (void)out_size; (void)ws_size;

    const float* x_t        = (const float*)d_in[0];
    const float* h_prev     = (const float*)d_in[1];
    const float* W_ih       = (const float*)d_in[2];
    const float* b_ih       = (const float*)d_in[3];
    const float* W_hh_fixed = (const float*)d_in[4];
    const float* W_hh_virt  = (const float*)d_in[5];
    const float* gumbel     = (const float*)d_in[6];

    float* out      = (float*)d_out;
    float* h_next   = out;                 // B*H
    float* W_new    = out + BB * HH;       // H*H

    float* ws            = (float*)d_ws;
    float* input_contrib = ws;             // B*H floats
    float* logits        = ws + BB * HH;   // H*H floats

    // K1: logits = fixed + virtual  (H*H/4 float4 elements)
    {
        int n4 = (HH * HH) / 4;
        k_add_logits<<<n4 / 256, 256, 0, stream>>>(W_hh_fixed, W_hh_virt, logits);
    }
    // K2: input_contrib = x_t @ W_ih^T + b_ih   (WMMA f32)
    {
        dim3 g(BB / 16, HH / 16);
        k_gemm_input<<<g, 32, 0, stream>>>(x_t, W_ih, b_ih, input_contrib);
    }
    // K3: fused gumbel-softmax exchange + tanh -> h_next
    {
        int nblocks = (BB * HH) / 8;       // 8 rows (waves) per block
        k_fused_exchange<<<nblocks, 256, 0, stream>>>(gumbel, h_prev, logits,
                                                      input_contrib, h_next);
    }
    // K4: hebbian WMMA + fast-weight update -> W_new
    {
        dim3 g(HH / 16, HH / 16);
        k_gemm_hebbian<<<g, 32, 0, stream>>>(h_next, h_prev, W_hh_virt, W_new);
    }
}